// MoeRouter_22153441313343
// MI455X (gfx1250) — compile-verified
//
#include <hip/hip_runtime.h>
#include <hip/hip_bf16.h>

typedef __attribute__((ext_vector_type(2))) float v2f;
typedef __attribute__((ext_vector_type(4))) float v4f;
typedef __attribute__((ext_vector_type(8))) float v8f;

#define HID      2048
#define NE       8
#define NTOK     16384
#define TOPK     2
#define CHUNK    64              // K elements staged per pipeline stage
#define NCHUNK   (HID / CHUNK)   // 32
#define XSTRIDE  68              // padded LDS row stride (dwords): bank = (4r+c)%64, conflict-free

// ---------------------------------------------------------------------------
// Kernel 1: router logits  [T,8] = x[T,2048] @ gate_w[8,2048]^T + gate_b
// One wave computes a 16-token tile with V_WMMA_F32_16X16X4_F32 (512 MACs).
// 256 blocks x 128 threads (4 waves) -> 64 tokens/block.
// ---------------------------------------------------------------------------
__global__ __launch_bounds__(128) void moe_router_logits_kernel(
    const float* __restrict__ x,
    const float* __restrict__ gate_w,
    const float* __restrict__ gate_b,
    float* __restrict__ logits)
{
    __shared__ float lds_w[HID * NE];            // transposed gate_w: wT[h][e], 64 KB
    __shared__ float lds_x[4][16 * XSTRIDE];     // per-wave x tile staging, 17 KB

    const int tid = threadIdx.x;

    // ---- Stage gate_w transposed into LDS (coalesced global float4 reads) ----
    for (int i = tid; i < (NE * HID) / 4; i += 128) {
        v4f g = ((const v4f*)gate_w)[i];
        int base = i * 4;
        int e  = base >> 11;          // / 2048
        int hh = base & (HID - 1);
        lds_w[(hh + 0) * NE + e] = g.x;
        lds_w[(hh + 1) * NE + e] = g.y;
        lds_w[(hh + 2) * NE + e] = g.z;
        lds_w[(hh + 3) * NE + e] = g.w;
    }

    const int wave = tid >> 5;
    const int lane = tid & 31;
    const int n    = lane & 15;       // column within tile (token-row for A, expert for B/D)
    const int h    = lane >> 4;       // lane half
    const int base_t = blockIdx.x * 64 + wave * 16;

    const float* xw = x + (size_t)base_t * HID;
    float*       lx = &lds_x[wave][0];

    // ---- Prefetch chunk 0 of x into registers (overlaps with gate_w staging) ----
    v4f stage[8];
#pragma unroll
    for (int i = 0; i < 8; ++i)
        stage[i] = *(const v4f*)(xw + (size_t)(2 * i + h) * HID + n * 4);

    __syncthreads();   // lds_w ready

    v8f acc = {};      // 16x16 f32 accumulator (tokens x experts)

    for (int c = 0; c < NCHUNK; ++c) {
        // stage registers -> LDS (per-wave region, no barrier needed; DS is in-order)
#pragma unroll
        for (int i = 0; i < 8; ++i)
            *(v4f*)(lx + (2 * i + h) * XSTRIDE + n * 4) = stage[i];

        // prefetch next chunk from global while WMMAs run
        if (c + 1 < NCHUNK) {
            const int kc = (c + 1) * CHUNK;
#pragma unroll
            for (int i = 0; i < 8; ++i)
                stage[i] = *(const v4f*)(xw + (size_t)(2 * i + h) * HID + kc + n * 4);
        }

        const int kc = c * CHUNK;
#pragma unroll
        for (int s = 0; s < CHUNK / 4; ++s) {
            // A fragment: lane(n,h) holds x[base_t+n][k+2h], x[...][k+2h+1]
            v2f a = *(const v2f*)(lx + n * XSTRIDE + 4 * s + 2 * h);

            // B fragment: VGPR0 = K row (0|2), VGPR1 = K row (1|3); col = n.
            // Lanes n>=8 read broadcast duplicates (conflict-free) then select 0.
            const int kb = kc + 4 * s + 2 * h;
            float b0 = lds_w[(kb + 0) * NE + (n & 7)];
            float b1 = lds_w[(kb + 1) * NE + (n & 7)];
            v2f b;
            b.x = (n < 8) ? b0 : 0.0f;
            b.y = (n < 8) ? b1 : 0.0f;

            acc = __builtin_amdgcn_wmma_f32_16x16x4_f32(
                /*neg_a=*/false, a, /*neg_b=*/false, b,
                /*c_mod=*/(short)0, acc, /*reuse_a=*/false, /*reuse_b=*/false);
        }
    }

    // ---- Store logits (+bias). D layout: acc[j] = D[j + 8h][n]. Only n<8 valid. ----
    if (n < 8) {
        const float bias = gate_b[n];
#pragma unroll
        for (int j = 0; j < 8; ++j) {
            const int row = base_t + j + 8 * h;
            logits[row * NE + n] = acc[j] + bias;
        }
    }
}

// ---------------------------------------------------------------------------
// Kernel 2: per-token softmax -> top-2 -> normalized weights, indices, one-hot
// mask transposed to [E, K, T]. One thread per token.
// ---------------------------------------------------------------------------
__global__ __launch_bounds__(256) void moe_router_epilogue_kernel(
    const float* __restrict__ logits,
    float* __restrict__ weights,   // [T, 2]
    float* __restrict__ idx_out,   // [T, 2]  (indices written as float values)
    float* __restrict__ mask_out)  // [E, K, T] (0.0/1.0)
{
    const int t = blockIdx.x * 256 + threadIdx.x;
    if (t >= NTOK) return;

    v4f l0 = *(const v4f*)(logits + (size_t)t * NE);
    v4f l1 = *(const v4f*)(logits + (size_t)t * NE + 4);
    float l[NE] = { l0.x, l0.y, l0.z, l0.w, l1.x, l1.y, l1.z, l1.w };

    float m = l[0];
#pragma unroll
    for (int e = 1; e < NE; ++e) m = fmaxf(m, l[e]);

    float p[NE];
#pragma unroll
    for (int e = 0; e < NE; ++e) p[e] = __expf(l[e] - m);

    // top-1 (strict > keeps lowest index on ties, matching jax.lax.top_k)
    int i0 = 0; float v0 = p[0];
#pragma unroll
    for (int e = 1; e < NE; ++e) { if (p[e] > v0) { v0 = p[e]; i0 = e; } }
    // top-2
    int i1 = -1; float v1 = -1.0f;
#pragma unroll
    for (int e = 0; e < NE; ++e) { if (e != i0 && p[e] > v1) { v1 = p[e]; i1 = e; } }

    // weights = topk(softmax) renormalized == p / (p0 + p1)  (softmax scale cancels)
    const float inv = 1.0f / (v0 + v1);
    weights[t * 2 + 0] = v0 * inv;
    weights[t * 2 + 1] = v1 * inv;

    idx_out[t * 2 + 0] = (float)i0;
    idx_out[t * 2 + 1] = (float)i1;

    // expert_mask[e][k][t] = (selected[t][k] == e); coalesced per (e,k) plane
#pragma unroll
    for (int e = 0; e < NE; ++e) {
        mask_out[(e * TOPK + 0) * NTOK + t] = (e == i0) ? 1.0f : 0.0f;
        mask_out[(e * TOPK + 1) * NTOK + t] = (e == i1) ? 1.0f : 0.0f;
    }
}

// ---------------------------------------------------------------------------
extern "C" void kernel_launch(void* const* d_in, const int* in_sizes, int n_in,
                              void* d_out, int out_size, void* d_ws, size_t ws_size,
                              hipStream_t stream)
{
    const float* x      = (const float*)d_in[0];   // [16384, 2048]
    const float* gate_w = (const float*)d_in[1];   // [8, 2048]
    const float* gate_b = (const float*)d_in[2];   // [8]

    float* out     = (float*)d_out;
    float* logits  = out;                              // 16384*8
    float* weights = logits  + (size_t)NTOK * NE;      // 16384*2
    float* idxs    = weights + (size_t)NTOK * TOPK;    // 16384*2
    float* mask    = idxs    + (size_t)NTOK * TOPK;    // 8*2*16384

    moe_router_logits_kernel<<<NTOK / 64, 128, 0, stream>>>(x, gate_w, gate_b, logits);
    moe_router_epilogue_kernel<<<NTOK / 256, 256, 0, stream>>>(logits, weights, idxs, mask);
}